// SpecialStackLayer_4045859193032
// MI455X (gfx1250) — compile-verified
//
#include <hip/hip_runtime.h>
#include <hip/hip_bf16.h>
#include <stdint.h>

// Problem constants (from reference): out[b,s,:] = hidden[b, pos[b,s], :]
#define BATCH 32
#define SEQ   4096
#define SENT  128
#define DIM   512

#define ROW_BYTES       (DIM * 4)          // 2048 B per gathered row
#define WAVES_PER_BLOCK 8                  // 256 threads, wave32
#define ROWS_PER_WAVE   2
#define ROWS_PER_BLOCK  (WAVES_PER_BLOCK * ROWS_PER_WAVE)     // 16
#define TOTAL_ROWS      (BATCH * SENT)                        // 4096
#define NUM_BLOCKS      (TOTAL_ROWS / ROWS_PER_BLOCK)         // 256
#define CHUNK_BYTES     (32 * 16)          // one wave, B128 per lane = 512 B
#define CHUNKS_PER_ROW  (ROW_BYTES / CHUNK_BYTES)             // 4

// Pure-gather copy using CDNA5 async global<->LDS DMA path (ASYNCcnt-tracked).
// Data never passes through VGPRs: global -> LDS (async) -> global (async).
__global__ __launch_bounds__(256)
void gather_rows_async_kernel(const float* __restrict__ hidden,
                              const int* __restrict__ pos,
                              float* __restrict__ out)
{
    __shared__ char smem[ROWS_PER_BLOCK * ROW_BYTES];   // 32 KB staging

    const int tid  = (int)threadIdx.x;
    const int wave = tid >> 5;
    const int lane = tid & 31;

    // Low 32 bits of a generic LDS pointer are the LDS byte offset
    // (flat LDS aperture puts the offset in addr[31:0]).
    const unsigned lds_base = (unsigned)(size_t)(&smem[0]);

    const uint64_t hidden_base = (uint64_t)(uintptr_t)hidden;
    const uint64_t out_base    = (uint64_t)(uintptr_t)out;

    // ---- Issue all async loads for this wave's rows (8 outstanding) ----
    #pragma unroll
    for (int r = 0; r < ROWS_PER_WAVE; ++r) {
        const int row = (int)blockIdx.x * ROWS_PER_BLOCK + wave * ROWS_PER_WAVE + r;
        const int b   = row >> 7;            // row / SENT
        const int p   = pos[row];            // gathered sequence position
        const uint64_t src_row = hidden_base
            + ((uint64_t)b * SEQ + (uint64_t)p) * (uint64_t)ROW_BYTES;
        const unsigned lds_row = lds_base
            + (unsigned)((wave * ROWS_PER_WAVE + r) * ROW_BYTES);

        #pragma unroll
        for (int k = 0; k < CHUNKS_PER_ROW; ++k) {
            const uint64_t gaddr = src_row + (uint64_t)(k * CHUNK_BYTES + lane * 16);
            const unsigned laddr = lds_row + (unsigned)(k * CHUNK_BYTES + lane * 16);
            asm volatile("global_load_async_to_lds_b128 %0, %1, off"
                         :
                         : "v"(laddr), "v"(gaddr)
                         : "memory");
        }
    }

    // Wait until this wave's async loads have landed in LDS.
    // (Each wave touches only its own LDS slice -> no workgroup barrier needed.)
    asm volatile("s_wait_asynccnt 0" ::: "memory");

    // ---- Issue async stores LDS -> out (drained implicitly by S_ENDPGM) ----
    #pragma unroll
    for (int r = 0; r < ROWS_PER_WAVE; ++r) {
        const int row = (int)blockIdx.x * ROWS_PER_BLOCK + wave * ROWS_PER_WAVE + r;
        const uint64_t dst_row = out_base + (uint64_t)row * (uint64_t)ROW_BYTES;
        const unsigned lds_row = lds_base
            + (unsigned)((wave * ROWS_PER_WAVE + r) * ROW_BYTES);

        #pragma unroll
        for (int k = 0; k < CHUNKS_PER_ROW; ++k) {
            const uint64_t gaddr = dst_row + (uint64_t)(k * CHUNK_BYTES + lane * 16);
            const unsigned laddr = lds_row + (unsigned)(k * CHUNK_BYTES + lane * 16);
            asm volatile("global_store_async_from_lds_b128 %0, %1, off"
                         :
                         : "v"(gaddr), "v"(laddr)
                         : "memory");
        }
    }
}

extern "C" void kernel_launch(void* const* d_in, const int* in_sizes, int n_in,
                              void* d_out, int out_size, void* d_ws, size_t ws_size,
                              hipStream_t stream) {
    (void)in_sizes; (void)n_in; (void)out_size; (void)d_ws; (void)ws_size;
    const float* hidden = (const float*)d_in[0];   // (BATCH, SEQ, DIM) f32
    const int*   pos    = (const int*)d_in[1];     // (BATCH, SENT) integer indices
    float*       out    = (float*)d_out;           // (BATCH, SENT, DIM) f32

    gather_rows_async_kernel<<<NUM_BLOCKS, 256, 0, stream>>>(hidden, pos, out);
}